// LlamaAttention_24189255811376
// MI455X (gfx1250) — compile-verified
//
#include <hip/hip_runtime.h>
#include <math.h>

// ---------------------------------------------------------------------------
// Llama decode attention (B=16, S=1, DIM=4096, NH=32, NKV=8, HD=128,
// kv_len = 4097) for MI455X / gfx1250.  f32 end-to-end.
//  * V_WMMA_F32_16X16X4_F32 for all matrix math (K remapped so all global
//    traffic is b128 per lane).
//  * KV cache streamed into LDS with GLOBAL_LOAD_ASYNC_TO_LDS_B128
//    (ASYNCcnt); phase-2 V tiles double-buffered with s_wait_asynccnt 4.
//  * Register double-buffered GEMM mainloop (partial loadcnt waits).
// ---------------------------------------------------------------------------

typedef float v2f __attribute__((ext_vector_type(2)));
typedef float v4f __attribute__((ext_vector_type(4)));
typedef float v8f __attribute__((ext_vector_type(8)));

#define NEG_INF (-__builtin_inff())

#if defined(__has_builtin)
#if __has_builtin(__builtin_amdgcn_s_wait_asynccnt)
#define WAIT_ASYNC_LE(n) __builtin_amdgcn_s_wait_asynccnt(n)
#endif
#endif
#ifndef WAIT_ASYNC_LE
#define WAIT_ASYNC_LE(n) asm volatile("s_wait_asynccnt %0" ::"i"(n) : "memory")
#endif

constexpr int kB      = 16;
constexpr int kDIM    = 4096;
constexpr int kNH     = 32;
constexpr int kNKV    = 8;
constexpr int kHD     = 128;
constexpr int kMAXLEN = 8192;
constexpr int kSPOS   = 4096;           // start_pos (matches reference const)
constexpr int kKVLEN  = kSPOS + 1;      // 4097
constexpr int kKVPAD  = 4112;           // 257 * 16
constexpr int kNREP   = kNH / kNKV;     // 4
constexpr int kROWP   = 132;            // LDS row stride in floats (128 + pad)

// workspace layout (float offsets)
constexpr size_t XQ_OFF   = 0;
constexpr size_t XK_OFF   = XQ_OFF + (size_t)kB * kDIM;
constexpr size_t XV_OFF   = XK_OFF + (size_t)kB * kNKV * kHD;
constexpr size_t ATTN_OFF = XV_OFF + (size_t)kB * kNKV * kHD;
constexpr size_t SC_OFF   = ATTN_OFF + (size_t)kB * kDIM;

static __device__ inline v8f wmma_f32(v2f a, v2f b, v8f c) {
  return __builtin_amdgcn_wmma_f32_16x16x4_f32(
      false, a, false, b, (short)0, c, false, false);
}

// One wave stages one 512B row (128 f32) into LDS: lane l moves 16 bytes.
static __device__ inline void async_row(const float* gsrc, unsigned lds_byte,
                                        int lane) {
  unsigned long long ga =
      (unsigned long long)(size_t)gsrc + 16ull * (unsigned)lane;
  unsigned ldsa = lds_byte + (unsigned)(lane << 4);
  asm volatile("global_load_async_to_lds_b128 %0, %1, off"
               :: "v"(ldsa), "v"(ga)
               : "memory");
}

// ---------------------------------------------------------------------------
// C[16 x N] = X[16 x K] * W[N x K]^T.  One 16-wide N tile per block, 4 waves
// split K.  K remap: lane loads float4 at k0+4*half; wmma#0 eats .xy,
// wmma#1 eats .zw (A and B agree, reduction order-invariant).
// Register double-buffered so next loads overlap the current WMMA pair.
// ---------------------------------------------------------------------------
__global__ __launch_bounds__(128) void gemm16_wmma(
    const float* __restrict__ X, const float* __restrict__ W,
    float* __restrict__ C, int K, int N) {
  const int tid  = threadIdx.x;
  const int wid  = __builtin_amdgcn_readfirstlane(tid >> 5);  // scalar
  const int l    = tid & 31;
  const int half = l >> 4;
  const int lm   = l & 15;
  const int n0   = blockIdx.x << 4;

  const size_t wrow = (size_t)(n0 + lm) * (size_t)K;
  const size_t xrow = (size_t)lm * (size_t)K;
  const int kchunk = K >> 2;          // per-wave K extent (multiple of 8)
  const int kbeg = wid * kchunk;      // scalar
  const int kend = kbeg + kchunk;     // scalar

  v8f acc = {0.f, 0.f, 0.f, 0.f, 0.f, 0.f, 0.f, 0.f};
  v4f an = *(const v4f*)(X + xrow + kbeg + (half << 2));
  v4f bn = *(const v4f*)(W + wrow + kbeg + (half << 2));
#pragma unroll 2
  for (int k0 = kbeg; k0 < kend - 8; k0 += 8) {
    const v4f ac = an, bc = bn;
    const int kb = k0 + 8 + (half << 2);
    if (((k0 + 8) & 255) == 0)  // scalar condition
      __builtin_prefetch(W + wrow + kb + 1024, 0, 3);
    an = *(const v4f*)(X + xrow + kb);   // global_load_b128 (next iter)
    bn = *(const v4f*)(W + wrow + kb);
    v2f a0, a1, b0, b1;
    a0.x = ac.x; a0.y = ac.y; a1.x = ac.z; a1.y = ac.w;
    b0.x = bc.x; b0.y = bc.y; b1.x = bc.z; b1.y = bc.w;
    acc = wmma_f32(a0, b0, acc);
    acc = wmma_f32(a1, b1, acc);
  }
  {  // epilogue
    v2f a0, a1, b0, b1;
    a0.x = an.x; a0.y = an.y; a1.x = an.z; a1.y = an.w;
    b0.x = bn.x; b0.y = bn.y; b1.x = bn.z; b1.y = bn.w;
    acc = wmma_f32(a0, b0, acc);
    acc = wmma_f32(a1, b1, acc);
  }

  __shared__ float red[4][32][8];
#pragma unroll
  for (int r = 0; r < 8; ++r) red[wid][l][r] = acc[r];
  __syncthreads();
  if (wid == 0) {
#pragma unroll
    for (int r = 0; r < 8; ++r) {
      float s = red[0][l][r] + red[1][l][r] + red[2][l][r] + red[3][l][r];
      const int m = r + (half << 3);  // C/D layout: M = r + 8*half
      C[(size_t)m * N + n0 + lm] = s;
    }
  }
}

// ---------------------------------------------------------------------------
// RoPE on xq (16*32 heads) and xk (16*8 heads), in place in workspace.
// ---------------------------------------------------------------------------
__global__ __launch_bounds__(256) void rope_kernel(
    float* __restrict__ xq, float* __restrict__ xk,
    const float* __restrict__ cosv, const float* __restrict__ sinv) {
  const int idx    = blockIdx.x * blockDim.x + threadIdx.x;
  const int qpairs = kB * kNH * (kHD / 2);   // 32768
  const int kpairs = kB * kNKV * (kHD / 2);  // 8192
  float* p;
  int i;
  if (idx < qpairs) {
    i = idx & 63;
    p = xq + (size_t)(idx >> 6) * kHD + (i << 1);
  } else if (idx < qpairs + kpairs) {
    const int t = idx - qpairs;
    i = t & 63;
    p = xk + (size_t)(t >> 6) * kHD + (i << 1);
  } else {
    return;
  }
  const float c = cosv[i], s = sinv[i];
  const float x0 = p[0], x1 = p[1];
  p[0] = x0 * c - x1 * s;
  p[1] = x0 * s + x1 * c;
}

// ---------------------------------------------------------------------------
// Attention: one block (4 waves) per (b, kv_head).  4 grouped query heads
// padded to an M=16 WMMA tile.
// Phase 1: K tiles DMA'd to LDS (async b128), Q hoisted to registers,
//          scores = Q*K^T -> ws + row max.
// Phase 2: V chunks DMA'd to double-buffered LDS (s_wait_asynccnt 4),
//          p = exp(s-max) as A, V as B, 8 accumulator tiles.
// ---------------------------------------------------------------------------
__global__ __launch_bounds__(128) void attn_kernel(
    const float* __restrict__ xq, const float* __restrict__ xk,
    const float* __restrict__ xv, const float* __restrict__ cache_k,
    const float* __restrict__ cache_v, float* __restrict__ scores,
    float* __restrict__ attn) {
  const int bid = blockIdx.x;  // b * NKV + kvh
  const int b   = bid >> 3;
  const int kvh = bid & 7;
  const int tid  = threadIdx.x;
  const int wid  = __builtin_amdgcn_readfirstlane(tid >> 5);  // scalar
  const int l    = tid & 31;
  const int half = l >> 4;
  const int lm   = l & 15;

  __shared__ __align__(16) float ktile[4][16 * kROWP];     // 33792 B
  __shared__ __align__(16) float vtile[4][2][4 * kROWP];   // 16896 B
  __shared__ float redmax[4][16][4];
  __shared__ float rowMax[4];
  __shared__ float redsum[4][32];
  __shared__ float rowSum[4];
  __shared__ float outAcc[4][128];

  for (int i = tid; i < 4 * kHD; i += 128) outAcc[i >> 7][i & 127] = 0.f;

  const float scale = 0.08838834764831845f;  // 1/sqrt(128)
  float* sc = scores + (size_t)bid * 4 * kKVPAD;
  const float* qb   = xq + (size_t)b * kDIM + (size_t)(kvh * kNREP) * kHD;
  const float* kcb  = cache_k + (size_t)b * kMAXLEN * kNKV * kHD + (size_t)kvh * kHD;
  const float* vcb  = cache_v + (size_t)b * kMAXLEN * kNKV * kHD + (size_t)kvh * kHD;
  const float* knew = xk + (size_t)(b * kNKV + kvh) * kHD;
  const float* vnew = xv + (size_t)(b * kNKV + kvh) * kHD;
  const int kvstride = kNKV * kHD;

  const unsigned kt_u32 = (unsigned)(size_t)&ktile[wid][0];
  const float*   ktp    = &ktile[wid][lm * kROWP];
  const unsigned vt_u32 = (unsigned)(size_t)&vtile[wid][0][0];
  const float*   vtp    = &vtile[wid][0][0];
  const float*   qrow   = qb + (size_t)lm * kHD;

  // Hoist this lane's Q fragments (loop-invariant across all K tiles).
  v4f qreg[16];
#pragma unroll
  for (int i = 0; i < 16; ++i) {
    if (lm < 4) qreg[i] = *(const v4f*)(qrow + i * 8 + (half << 2));
    else { qreg[i].x = 0.f; qreg[i].y = 0.f; qreg[i].z = 0.f; qreg[i].w = 0.f; }
  }

  // ---- phase 1: scores ----------------------------------------------------
  float m0 = NEG_INF, m1 = NEG_INF, m2 = NEG_INF, m3 = NEG_INF;
  for (int t = wid; t < kKVPAD / 16; t += 4) {  // 257 tiles of 16 positions
#pragma unroll 4
    for (int r = 0; r < 16; ++r) {
      const int pos  = (t << 4) + r;
      const int posc = pos < kSPOS ? pos : kSPOS;
      const float* src =
          (posc < kSPOS) ? (kcb + (size_t)posc * kvstride) : knew;
      async_row(src, kt_u32 + (unsigned)(r * kROWP * 4), l);
    }
    WAIT_ASYNC_LE(0);

    v8f cacc = {0.f, 0.f, 0.f, 0.f, 0.f, 0.f, 0.f, 0.f};
#pragma unroll
    for (int i = 0; i < 16; ++i) {
      v4f b4 = *(const v4f*)(ktp + i * 8 + (half << 2));  // ds_load_b128
      v2f a0, a1, b0, b1;
      a0.x = qreg[i].x; a0.y = qreg[i].y; a1.x = qreg[i].z; a1.y = qreg[i].w;
      b0.x = b4.x;      b0.y = b4.y;      b1.x = b4.z;      b1.y = b4.w;
      cacc = wmma_f32(a0, b0, cacc);
      cacc = wmma_f32(a1, b1, cacc);
    }

    const int pos = (t << 4) + lm;
    if (l < 16) {  // rows 0..3 live in VGPRs 0..3 of lanes 0..15
      const bool valid = pos < kKVLEN;
      float s0 = valid ? cacc[0] * scale : NEG_INF;
      float s1 = valid ? cacc[1] * scale : NEG_INF;
      float s2 = valid ? cacc[2] * scale : NEG_INF;
      float s3 = valid ? cacc[3] * scale : NEG_INF;
      m0 = fmaxf(m0, s0); m1 = fmaxf(m1, s1);
      m2 = fmaxf(m2, s2); m3 = fmaxf(m3, s3);
      sc[0 * kKVPAD + pos] = s0;
      sc[1 * kKVPAD + pos] = s1;
      sc[2 * kKVPAD + pos] = s2;
      sc[3 * kKVPAD + pos] = s3;
    }
  }
  if (l < 16) {
    redmax[wid][lm][0] = m0; redmax[wid][lm][1] = m1;
    redmax[wid][lm][2] = m2; redmax[wid][lm][3] = m3;
  }
  __threadfence();  // score stores visible block-wide
  __syncthreads();
  if (tid < 4) {
    float mm = NEG_INF;
    for (int w = 0; w < 4; ++w)
      for (int i = 0; i < 16; ++i) mm = fmaxf(mm, redmax[w][i][tid]);
    rowMax[tid] = mm;
  }
  __syncthreads();

  // ---- phase 2: out = softmax(scores) * V ---------------------------------
  const float rmj = (lm < 4) ? rowMax[lm] : 0.f;
  float lsum = 0.f;
  v8f o[8];
#pragma unroll
  for (int t = 0; t < 8; ++t)
    o[t] = (v8f){0.f, 0.f, 0.f, 0.f, 0.f, 0.f, 0.f, 0.f};

  constexpr unsigned kVbuf = 4u * kROWP * 4u;  // bytes per V buffer
  // stage chunk -> LDS buffer (4 async rows)
  auto stageV = [&](int chunk, int buf) {
    const int p0 = chunk << 2;
#pragma unroll
    for (int r = 0; r < 4; ++r) {
      int pr = p0 + r; if (pr > kSPOS) pr = kSPOS;
      const float* src = (pr < kSPOS) ? (vcb + (size_t)pr * kvstride) : vnew;
      async_row(src, vt_u32 + (unsigned)buf * kVbuf + (unsigned)(r * kROWP * 4), l);
    }
  };

  stageV(wid, 0);  // prologue: first chunk into buffer 0
  int cur = 0;
  for (int ch = wid; ch < kKVPAD / 4; ch += 4) {  // 1028 chunks of 4 pos
    // stage next chunk into the other buffer (clamped re-stage on last iter
    // keeps the ASYNCcnt accounting uniform; data unused).
    const int chn = (ch + 4 < kKVPAD / 4) ? (ch + 4) : ch;
    stageV(chn, cur ^ 1);

    const int kb = (ch << 2) + (half << 1);
    v2f a;
    if (lm < 4) {
      v2f sv = *(const v2f*)(sc + (size_t)lm * kKVPAD + kb);
      a.x = __expf(sv.x - rmj);
      a.y = __expf(sv.y - rmj);
    } else {
      a.x = 0.f; a.y = 0.f;
    }
    lsum += a.x + a.y;

    WAIT_ASYNC_LE(4);  // current buffer done; next 4 still in flight
    const float* vb = vtp + (unsigned)cur * (4 * kROWP);
    const int r0 = (half << 1) * kROWP;  // b.x row = p0+2h, b.y row = +1
#pragma unroll
    for (int t = 0; t < 8; ++t) {
      const int d = (t << 4) + lm;
      v2f bb;
      bb.x = vb[r0 + d];            // ds_load_b32
      bb.y = vb[r0 + kROWP + d];    // ds_load_b32
      o[t] = wmma_f32(a, bb, o[t]);
    }
    cur ^= 1;
  }
  WAIT_ASYNC_LE(0);  // drain the speculative last stage

  redsum[wid][l] = lsum;
  __syncthreads();
  if (tid < 4) {
    float ss = 0.f;
    for (int w = 0; w < 4; ++w) ss += redsum[w][tid] + redsum[w][tid + 16];
    rowSum[tid] = ss;
  }
  __syncthreads();

  if (l < 16) {
#pragma unroll
    for (int t = 0; t < 8; ++t)
#pragma unroll
      for (int r = 0; r < 4; ++r)
        atomicAdd(&outAcc[r][(t << 4) + lm], o[t][r]);  // ds_add_f32
  }
  __syncthreads();

  for (int i = tid; i < 4 * kHD; i += 128) {
    const int r = i >> 7, d = i & 127;
    attn[(size_t)b * kDIM + (size_t)(kvh * kNREP + r) * kHD + d] =
        outAcc[r][d] / rowSum[r];
  }
}

// ---------------------------------------------------------------------------
extern "C" void kernel_launch(void* const* d_in, const int* in_sizes, int n_in,
                              void* d_out, int out_size, void* d_ws,
                              size_t ws_size, hipStream_t stream) {
  (void)in_sizes; (void)n_in; (void)out_size; (void)ws_size;
  const float* x    = (const float*)d_in[0];
  // d_in[1] = start_pos (int) — fixed at 4096, matching the reference const.
  const float* fcos = (const float*)d_in[2];
  const float* fsin = (const float*)d_in[3];
  const float* ck   = (const float*)d_in[4];
  const float* cv   = (const float*)d_in[5];
  const float* wq   = (const float*)d_in[6];
  const float* wk   = (const float*)d_in[7];
  const float* wv   = (const float*)d_in[8];
  const float* wo   = (const float*)d_in[9];

  float* ws   = (float*)d_ws;
  float* xq   = ws + XQ_OFF;
  float* xk   = ws + XK_OFF;
  float* xv   = ws + XV_OFF;
  float* attn = ws + ATTN_OFF;
  float* sc   = ws + SC_OFF;

  gemm16_wmma<<<dim3(kDIM / 16), 128, 0, stream>>>(x, wq, xq, kDIM, kDIM);
  gemm16_wmma<<<dim3(kNKV * kHD / 16), 128, 0, stream>>>(x, wk, xk, kDIM,
                                                         kNKV * kHD);
  gemm16_wmma<<<dim3(kNKV * kHD / 16), 128, 0, stream>>>(x, wv, xv, kDIM,
                                                         kNKV * kHD);
  rope_kernel<<<dim3(160), 256, 0, stream>>>(xq, xk, fcos, fsin);
  attn_kernel<<<dim3(kB * kNKV), 128, 0, stream>>>(xq, xk, xv, ck, cv, sc,
                                                   attn);
  gemm16_wmma<<<dim3(kDIM / 16), 128, 0, stream>>>(attn, wo, (float*)d_out,
                                                   kDIM, kDIM);
}